// kernel_generated_0_62904091017234
// MI455X (gfx1250) — compile-verified
//
#include <hip/hip_runtime.h>

typedef __attribute__((ext_vector_type(2))) float v2f;
typedef __attribute__((ext_vector_type(8))) float v8f;

// One workgroup per (b, o): fused channel-mix + double-unfold conv + roll.
// out[b, j*4+p, (o+1)%56, w] =
//   sum_{m,kA,kB} w0[m,kB,kA,j] * t4[b,m,o, w+kA+kB-2, p] * (pad validity)
// where t4[m,w,p] = sum_j x[b,2j+m,o,w] * w1[j,p].
__global__ __launch_bounds__(256) void fused_mix_unfold_conv(
    const float* __restrict__ x,   // (128,128,56,56)
    const float* __restrict__ w0,  // (2,3,3,32) = 576
    const float* __restrict__ w1,  // (64,4)     = 256
    float* __restrict__ out)       // (128,128,56,56)
{
  __shared__ float sx[128 * 56];   // x row slab [c][w]
  __shared__ float sw1[256];
  __shared__ float sw0[576];
  __shared__ float st4[2 * 72 * 4]; // [m][w''=w+2 (0..71)][p], zero halo

  const int t = threadIdx.x;
  const int o = blockIdx.x;   // H row of the pre-roll output
  const int b = blockIdx.y;

  // ---------------- Phase A: cooperative loads + zero t4 halo ----------------
  {
    const int c = t >> 1, half = t & 1;
    const float4* src =
        (const float4*)(x + (((size_t)b * 128 + c) * 56 + o) * 56) + half * 7;
    float4* dst = (float4*)(sx + c * 56) + half * 7;
#pragma unroll
    for (int i = 0; i < 7; ++i) dst[i] = src[i];
  }
  if (t < 64)  ((float4*)sw1)[t] = ((const float4*)w1)[t];
  if (t < 144) ((float4*)sw0)[t] = ((const float4*)w0)[t];
  for (int e = t; e < 576; e += 256) st4[e] = 0.0f;
  __syncthreads();

  // ---------------- Phase B: stage-1 channel mix (VALU, K=64) ----------------
  // t4[m][w][p] = sum_j x[2j+m][w] * w1[j][p]
  for (int e = t; e < 448; e += 256) {
    const int m = e / 224, r = e % 224;
    const int w = r >> 2, p = r & 3;
    const float* xr = sx + m * 56 + w;  // element j at offset j*112
    float acc = 0.0f;
#pragma unroll 8
    for (int j = 0; j < 64; ++j) acc += xr[j * 112] * sw1[j * 4 + p];
    st4[(m * 72 + (w + 2)) * 4 + p] = acc;
  }
  __syncthreads();

  // ---------------- Phase C: stage-2 GEMM via fp32 WMMA ----------------
  // Per p: out_p(56x32) = U5_p(56 x 12) @ W5(12 x 32),
  // K index kk = m*6 + s (s=0..4 shift, s=5 zero pad).
  const int lane  = t & 31;
  const int wv    = t >> 5;
  const int Mtile = wv >> 1;          // 0..3  (w tile)
  const int Ntile = wv & 1;           // 0..1  (j tile)
  const int ln    = lane & 15;
  const int hi    = lane >> 4;        // K half: lanes 0-15 -> K={k0,k0+1}, 16-31 -> K={k0+2,k0+3}
  const int jcol  = Ntile * 16 + ln;  // N = j
  const int wA    = Mtile * 16 + ln;  // M = w

  // B fragments: W5[m,s,j] = sum_{kA+kB=s} w0[m][kB][kA][j]  (shared over p)
  v2f bf[3];
#pragma unroll
  for (int ks = 0; ks < 3; ++ks) {
    float bv[2];
#pragma unroll
    for (int q = 0; q < 2; ++q) {
      const int kk = ks * 4 + hi * 2 + q;
      const int m = (kk >= 6) ? 1 : 0;
      const int s = kk - 6 * m;
      float v = 0.0f;
#pragma unroll
      for (int kA = 0; kA <= 2; ++kA) {
        const int kB = s - kA;
        if (s <= 4 && kB >= 0 && kB <= 2)
          v += sw0[((m * 3 + kB) * 3 + kA) * 32 + jcol];
      }
      bv[q] = v;
    }
    v2f bb; bb.x = bv[0]; bb.y = bv[1];
    bf[ks] = bb;
  }

  v8f acc[4];
#pragma unroll
  for (int p = 0; p < 4; ++p) acc[p] = (v8f){0, 0, 0, 0, 0, 0, 0, 0};

#pragma unroll
  for (int p = 0; p < 4; ++p) {
#pragma unroll
    for (int ks = 0; ks < 3; ++ks) {
      float av[2];
#pragma unroll
      for (int q = 0; q < 2; ++q) {
        const int kk = ks * 4 + hi * 2 + q;
        const int m = (kk >= 6) ? 1 : 0;
        const int s = kk - 6 * m;
        const float v = st4[(m * 72 + (wA + s)) * 4 + p];  // halo gives zeros
        av[q] = (s <= 4) ? v : 0.0f;
      }
      v2f a; a.x = av[0]; a.y = av[1];
      acc[p] = __builtin_amdgcn_wmma_f32_16x16x4_f32(
          false, a, false, bf[ks], (short)0, acc[p], false, false);
    }
  }

  // Boundary corrections: w=0 drops (kA=2,kB=0); w=55 drops (kA=0,kB=2).
  if (hi == 0) {
    if (Mtile == 0) {
#pragma unroll
      for (int p = 0; p < 4; ++p) {
        const float corr = sw0[2 * 32 + jcol]  * st4[(0 * 72 + 2) * 4 + p] +
                           sw0[11 * 32 + jcol] * st4[(1 * 72 + 2) * 4 + p];
        acc[p][0] -= corr;
      }
    }
    if (Mtile == 3) {
#pragma unroll
      for (int p = 0; p < 4; ++p) {
        const float corr = sw0[6 * 32 + jcol]  * st4[(0 * 72 + 57) * 4 + p] +
                           sw0[15 * 32 + jcol] * st4[(1 * 72 + 57) * 4 + p];
        acc[p][7] -= corr;
      }
    }
  }

  // Store with H-roll: D VGPR r -> w = Mtile*16 + hi*8 + r, N lane -> j.
  const int h = (o + 1) % 56;
  const int wbase = Mtile * 16 + hi * 8;
  if (wbase < 56) {  // drop padded M rows 56..63 (Mtile 3, upper half)
#pragma unroll
    for (int p = 0; p < 4; ++p) {
      const int c = jcol * 4 + p;
      float* dst = out + ((((size_t)b * 128 + c) * 56 + h) * 56 + wbase);
      float4 lo = make_float4(acc[p][0], acc[p][1], acc[p][2], acc[p][3]);
      float4 hi4 = make_float4(acc[p][4], acc[p][5], acc[p][6], acc[p][7]);
      ((float4*)dst)[0] = lo;
      ((float4*)dst)[1] = hi4;
    }
  }
}

extern "C" void kernel_launch(void* const* d_in, const int* in_sizes, int n_in,
                              void* d_out, int out_size, void* d_ws, size_t ws_size,
                              hipStream_t stream) {
  const float* x  = (const float*)d_in[0];
  const float* w0 = (const float*)d_in[1];
  const float* w1 = (const float*)d_in[2];
  float* out = (float*)d_out;
  dim3 grid(56, 128);  // (o, b)
  fused_mix_unfold_conv<<<grid, 256, 0, stream>>>(x, w0, w1, out);
}